// SolveScheduling_41850161332250
// MI455X (gfx1250) — compile-verified
//
#include <hip/hip_runtime.h>
#include <math.h>

// Problem constants (match reference)
#define NN 24            // variables per problem
#define NP 23            // constraint pairs (M = 2*NP)
#define MM 46            // ramp constraints
#define BLOCK 64         // threads per block (2 waves, wave32)
#define N_SQP_TOTAL 21   // 20 SQP steps + 1 final (forward-identical) QP
#define N_IPM 20
#define C_RAMP 0.4f
#define SIGMA_CENTER 0.1f

// Per-thread LDS scratch layout, stride = BLOCK (bank-conflict-free: lane -> bank)
#define OFF_MU 0          // mu[24]
#define OFF_RS 24         // 1/sig[24]  (async-loaded raw sig, fixed up in place)
#define OFF_PQ 48         // Thomas per-step {q0,q1,p0,p1} x 23 = 92
#define OFF_DL 140        // dlam[46]
#define OFF_RL 186        // 1/lam[46]
#define LDS_PER 232       // floats per thread -> 232*64*4 = 59392 B static LDS

// 1-ulp hardware reciprocal: used for the ~115 per-IPM-iteration reciprocals
// (1/lam, 1/s, 1/det). A 1-ulp Newton-step perturbation only perturbs the IPM
// search direction by ~1e-7 relative; the iteration contracts it away.
__device__ __forceinline__ float frcp_fast(float x) {
#if __has_builtin(__builtin_amdgcn_rcpf)
  return __builtin_amdgcn_rcpf(x);      // v_rcp_f32
#else
  return 1.0f / x;
#endif
}

// Refined reciprocal for "data" paths (1/sig, 1/Qd, final alpha).
__device__ __forceinline__ float frcp(float x) {
  float r = frcp_fast(x);
  return r * (2.0f - x * r);            // one Newton step -> ~0.5 ulp
}

// CDNA5 async global->LDS copy (ASYNCcnt path). Each lane supplies its own
// 64-bit global address and 32-bit LDS byte address; the async unit writes
// LDS directly (no VGPR round-trip), giving us the [element][lane] transpose
// for free.
__device__ __forceinline__ void async_g2l_b32(unsigned lds_byte_addr,
                                              unsigned long long gaddr) {
  asm volatile("global_load_async_to_lds_b32 %0, %1, off"
               :: "v"(lds_byte_addr), "v"(gaddr) : "memory");
}

__device__ __forceinline__ void wait_async0() {
#if __has_builtin(__builtin_amdgcn_s_wait_asynccnt)
  __builtin_amdgcn_s_wait_asynccnt(0);
#else
  asm volatile("s_wait_asynccnt 0x0" ::: "memory");
#endif
}

__global__ __launch_bounds__(BLOCK, 1) void sqp_ipm_kernel(
    const float* __restrict__ mu, const float* __restrict__ sig,
    float* __restrict__ out, int batch) {
  __shared__ float lds[LDS_PER * BLOCK];
  const int tid = threadIdx.x;
  const int gt  = blockIdx.x * BLOCK + tid;
  const bool active = gt < batch;
  const int bi = active ? gt : (batch - 1);
  const float* mup  = mu  + (size_t)bi * NN;
  const float* sigp = sig + (size_t)bi * NN;
  __builtin_prefetch(mup, 0, 0);   // global_prefetch_b8 (gfx1250)
  __builtin_prefetch(sigp, 0, 0);

  // ---- async-stage mu and sig into LDS (transposed), ASYNCcnt-tracked ----
#pragma unroll
  for (int j = 0; j < NN; ++j) {
    unsigned la_m = (unsigned)(uintptr_t)&lds[(OFF_MU + j) * BLOCK + tid];
    unsigned la_s = (unsigned)(uintptr_t)&lds[(OFF_RS + j) * BLOCK + tid];
    async_g2l_b32(la_m, (unsigned long long)(uintptr_t)(mup + j));
    async_g2l_b32(la_s, (unsigned long long)(uintptr_t)(sigp + j));
  }
  wait_async0();   // s_wait_asynccnt 0: our lane's columns are complete
                   // (lane consumes only its own column -> no barrier needed)

  float z[NN];
#pragma unroll
  for (int j = 0; j < NN; ++j) {           // fixup: z0 = mu ; RS <- 1/sig
    z[j] = lds[(OFF_MU + j) * BLOCK + tid];
    float sg = lds[(OFF_RS + j) * BLOCK + tid];
    lds[(OFF_RS + j) * BLOCK + tid] = frcp(sg);
  }

  float Qd[NN], p[NN], u[NN], lam[MM], s[MM];

#pragma unroll 1
  for (int sqp = 0; sqp < N_SQP_TOTAL; ++sqp) {
    // ---- Linearize: dG, d2G at current z; build diagonal QP (Qd, p) ----
#pragma unroll
    for (int j = 0; j < NN; ++j) {
      float m  = lds[(OFF_MU + j) * BLOCK + tid];
      float rs = lds[(OFF_RS + j) * BLOCK + tid];
      float tj = (z[j] - m) * rs;
      // (GU+GO)*0.5 = 25.25 ; GU = 50 ; (GU+GO)*inv_sqrt_2pi = 50.5*0.39894228
      float dg  = 25.25f * (1.0f + erff(tj * 0.70710678118654752f)) - 50.0f;
      float d2g = 50.5f * 0.3989422804014327f * __expf(-0.5f * tj * tj) * rs;
      Qd[j] = d2g + 1.0f;
      u[j]  = frcp(Qd[j]);
      p[j]  = dg - d2g * z[j] - m;
      z[j]  = 0.0f;                 // IPM starts at z=0
    }
#pragma unroll
    for (int i = 0; i < MM; ++i) { lam[i] = 1.0f; s[i] = 1.0f; }

    // ---- 20 primal-dual IPM iterations ----
#pragma unroll 1
    for (int it = 0; it < N_IPM; ++it) {
      // sigma * mu_gap
      float dot = 0.0f;
#pragma unroll
      for (int i = 0; i < MM; ++i) dot += lam[i] * s[i];
      const float smg = SIGMA_CENTER * dot * (1.0f / (float)MM);

      // ---- fused rhs build + block-Thomas forward sweep ----
      // K (constraint i interleaved with 23+i) is 2x2-block tridiagonal:
      //   B_i = [[Aii+f0, -Aii],[-Aii, Aii+f1]],  C_i = c_i * v v^T, v=(1,-1),
      //   Aii = u_i + u_{i+1}, c_i = -u_{i+1}, f = s/lam.
      float dl_prev = lam[0] - lam[NP];                       // dl_0
      float w_i = (Qd[0] * z[0] + p[0] + dl_prev) * u[0];     // w_0 = r1_0/Qd_0
      float cprev = 0.0f, beta = 0.0f, gsc = 0.0f;
#pragma unroll
      for (int i = 0; i < NP; ++i) {
        // w_{i+1}: r1_{i+1} = Qd z + p + (G^T lam)_{i+1}
        float dl_n, gtl;
        if (i + 1 < NP) { dl_n = lam[i + 1] - lam[i + 1 + NP]; gtl = dl_n - dl_prev; }
        else            { dl_n = 0.0f;                         gtl = -dl_prev; }
        float w_n = (Qd[i + 1] * z[i + 1] + p[i + 1] + gtl) * u[i + 1];

        float ti  = (z[i] - z[i + 1]) - (w_i - w_n);
        float rl0 = frcp_fast(lam[i]);
        float rl1 = frcp_fast(lam[i + NP]);
        lds[(OFF_RL + i)      * BLOCK + tid] = rl0;
        lds[(OFF_RL + i + NP) * BLOCK + tid] = rl1;
        float rhs0 =  ti - C_RAMP + smg * rl0;    // r2 - r3/lam - G(r1/Qd), simplified
        float rhs1 = -ti - C_RAMP + smg * rl1;

        float Aii = u[i] + u[i + 1];
        float e00 = Aii + s[i] * rl0;
        float e01 = -Aii;
        float e11 = Aii + s[i + NP] * rl1;
        if (i > 0) {                               // subtract C^T Btilde^{-1} C (rank-1)
          float tt = cprev * cprev * beta;
          e00 -= tt; e01 += tt; e11 -= tt;
          float cg = cprev * gsc;
          rhs0 -= cg; rhs1 += cg;
        }
        float rdet = frcp_fast(e00 * e11 - e01 * e01);  // SPD => det > 0
        float q0 = (e11 + e01) * rdet;             // q = Btilde^{-1} v
        float q1 = -(e01 + e00) * rdet;
        float p0 = (e11 * rhs0 - e01 * rhs1) * rdet;   // p = Btilde^{-1} btilde
        float p1 = (e00 * rhs1 - e01 * rhs0) * rdet;
        beta = q0 - q1;                            // v^T Btilde^{-1} v
        gsc  = p0 - p1;                            // v^T Btilde^{-1} btilde
        lds[(OFF_PQ + 4 * i + 0) * BLOCK + tid] = q0;
        lds[(OFF_PQ + 4 * i + 1) * BLOCK + tid] = q1;
        lds[(OFF_PQ + 4 * i + 2) * BLOCK + tid] = p0;
        lds[(OFF_PQ + 4 * i + 3) * BLOCK + tid] = p1;
        cprev = -u[i + 1];
        w_i = w_n; dl_prev = dl_n;
      }

      // ---- back substitution: dlam (interleaved pairs) ----
      float x0 = lds[(OFF_PQ + 4 * (NP - 1) + 2) * BLOCK + tid];
      float x1 = lds[(OFF_PQ + 4 * (NP - 1) + 3) * BLOCK + tid];
      lds[(OFF_DL + (NP - 1))      * BLOCK + tid] = x0;
      lds[(OFF_DL + (NP - 1) + NP) * BLOCK + tid] = x1;
#pragma unroll
      for (int i = NP - 2; i >= 0; --i) {
        float ci = -u[i + 1];
        float cv = ci * (x0 - x1);                 // c_i * (v . x_{i+1})
        float q0 = lds[(OFF_PQ + 4 * i + 0) * BLOCK + tid];
        float q1 = lds[(OFF_PQ + 4 * i + 1) * BLOCK + tid];
        float p0 = lds[(OFF_PQ + 4 * i + 2) * BLOCK + tid];
        float p1 = lds[(OFF_PQ + 4 * i + 3) * BLOCK + tid];
        x0 = p0 - cv * q0;
        x1 = p1 - cv * q1;
        lds[(OFF_DL + i)      * BLOCK + tid] = x0;
        lds[(OFF_DL + i + NP) * BLOCK + tid] = x1;
      }

      // ---- step length: alpha = 0.99 / max(1, max(-dlam/lam), max(-ds/s)) ----
      float mred = 1.0f;
#pragma unroll
      for (int i = 0; i < MM; ++i) {
        float dli = lds[(OFF_DL + i) * BLOCK + tid];
        float rli = lds[(OFF_RL + i) * BLOCK + tid];
        mred = fmaxf(mred, -dli * rli);
        float ds = -(s[i] - smg * rli) - s[i] * rli * dli;   // -(r3 + s*dlam)/lam
        mred = fmaxf(mred, -ds * frcp_fast(s[i]));
      }
      float alpha = 0.99f * frcp(mred);

      // ---- z update (uses OLD lam): dz = -(r1 + G^T dlam)/Qd ----
      float ddl_prev = 0.0f, dl_prev2 = 0.0f;
#pragma unroll
      for (int j = 0; j < NN; ++j) {
        float ddl_j = 0.0f, dl_j = 0.0f;
        if (j < NP) {
          ddl_j = lds[(OFF_DL + j) * BLOCK + tid] - lds[(OFF_DL + j + NP) * BLOCK + tid];
          dl_j  = lam[j] - lam[j + NP];
        }
        float gtd = ddl_j - ddl_prev;              // (G^T dlam)_j
        float gtl = dl_j - dl_prev2;               // (G^T lam)_j
        float r1j = Qd[j] * z[j] + p[j] + gtl;
        z[j] += alpha * (-(r1j + gtd) * u[j]);
        ddl_prev = ddl_j; dl_prev2 = dl_j;
      }

      // ---- lam, s update ----
#pragma unroll
      for (int i = 0; i < MM; ++i) {
        float dli = lds[(OFF_DL + i) * BLOCK + tid];
        float rli = lds[(OFF_RL + i) * BLOCK + tid];
        float ds  = -(s[i] - smg * rli) - s[i] * rli * dli;
        lam[i] += alpha * dli;
        s[i]   += alpha * ds;
      }
    } // IPM
  }   // SQP

  if (active) {
    float* op = out + (size_t)gt * NN;
#pragma unroll
    for (int j = 0; j < NN; ++j) op[j] = z[j];
  }
}

extern "C" void kernel_launch(void* const* d_in, const int* in_sizes, int n_in,
                              void* d_out, int out_size, void* d_ws, size_t ws_size,
                              hipStream_t stream) {
  (void)n_in; (void)out_size; (void)d_ws; (void)ws_size;
  const float* mu  = (const float*)d_in[0];
  const float* sig = (const float*)d_in[1];
  float* out = (float*)d_out;
  int batch = in_sizes[0] / NN;
  int grid  = (batch + BLOCK - 1) / BLOCK;
  sqp_ipm_kernel<<<grid, BLOCK, 0, stream>>>(mu, sig, out, batch);
}